// LengthRegulatorOld_7344394076707
// MI455X (gfx1250) — compile-verified
//
#include <hip/hip_runtime.h>
#include <hip/hip_bf16.h>
#include <stdint.h>

// Problem constants from the reference: x[16,2048,384] f32, dur[16,2048] i32.
#define LR_B 16
#define LR_T 2048
#define LR_C 384

// ---------------------------------------------------------------------------
// CDNA5 async global->LDS staging helpers (ASYNCcnt-tracked path).
// Builtin signature (from clang diagnostic): first param is
//   int __vector(4) addrspace(1)*  -- global source
// second is the LDS destination.
// ---------------------------------------------------------------------------
typedef int lr_v4i __attribute__((vector_size(16)));
typedef __attribute__((address_space(1))) lr_v4i* lr_gv4p;
typedef __attribute__((address_space(3))) lr_v4i* lr_lv4p;

__device__ __forceinline__ void lr_async_ld_b128(const int* g, int* lds_generic) {
#if defined(__has_builtin) && __has_builtin(__builtin_amdgcn_global_load_async_to_lds_b128)
  __builtin_amdgcn_global_load_async_to_lds_b128(
      (lr_gv4p)g, (lr_lv4p)lds_generic, /*offset=*/0, /*cpol=*/0);
#else
  // ISA 15.18.3 opcode 98: GLOBAL_LOAD_ASYNC_TO_LDS_B128, GV mode (64-bit vaddr).
  uint32_t lds_addr = (uint32_t)(uintptr_t)lds_generic;
  asm volatile("global_load_async_to_lds_b128 %0, %1, off"
               :: "v"(lds_addr), "v"(g)
               : "memory");
#endif
}

__device__ __forceinline__ void lr_wait_async0() {
#if defined(__has_builtin) && __has_builtin(__builtin_amdgcn_s_wait_asynccnt)
  __builtin_amdgcn_s_wait_asynccnt(0);
#else
  asm volatile("s_wait_asynccnt 0" ::: "memory");
#endif
}

// ---------------------------------------------------------------------------
// Kernel 1: per-batch inclusive cumsum of max(dur,0). One block per batch row.
// 256 threads * 8 elements = 2048.
// ---------------------------------------------------------------------------
__global__ __launch_bounds__(256)
void lr_cumsum_kernel(const int* __restrict__ dur, int* __restrict__ cs) {
  __shared__ int s_part[256];
  const int b = blockIdx.x;
  const int tid = threadIdx.x;
  const int* drow = dur + (size_t)b * LR_T;
  int* crow = cs + (size_t)b * LR_T;

  int loc[8];
  int run = 0;
  const int base = tid * 8;
#pragma unroll
  for (int k = 0; k < 8; ++k) {
    int d = drow[base + k];
    d = (d < 0) ? 0 : d;
    run += d;
    loc[k] = run;
  }
  s_part[tid] = run;
  __syncthreads();

  // Hillis-Steele inclusive scan over the 256 per-thread partial sums.
#pragma unroll
  for (int off = 1; off < 256; off <<= 1) {
    int add = (tid >= off) ? s_part[tid - off] : 0;
    __syncthreads();
    s_part[tid] += add;
    __syncthreads();
  }
  const int excl = s_part[tid] - run;  // exclusive prefix for this thread

#pragma unroll
  for (int k = 0; k < 8; ++k)
    crow[base + k] = loc[k] + excl;
}

// ---------------------------------------------------------------------------
// Kernel 2: expansion. Block = 256 threads (8 waves), 8 frames per wave.
// LDS holds this batch row's 8KB cumsum, staged with async b128 loads.
// ---------------------------------------------------------------------------
#define FRAMES_PER_WAVE 8
#define FRAMES_PER_BLOCK 64  // 8 waves * 8 frames

__global__ __launch_bounds__(256)
void lr_expand_kernel(const float* __restrict__ x, const int* __restrict__ cs,
                      float* __restrict__ out, int M) {
  __shared__ int s_cs[LR_T];  // 8 KB

  const int b = blockIdx.y;
  const int frame0 = blockIdx.x * FRAMES_PER_BLOCK;
  const int tid = threadIdx.x;
  const int lane = tid & 31;
  const int wave = tid >> 5;

  // Stage cs[b][0..T-1] into LDS: 512 b128 async transfers across 256 threads.
  const int* csrow = cs + (size_t)b * LR_T;
#pragma unroll
  for (int i = tid; i < LR_T / 4; i += 256)
    lr_async_ld_b128(csrow + i * 4, &s_cs[i * 4]);
  lr_wait_async0();
  __syncthreads();

  const int total = s_cs[LR_T - 1];

  for (int f = 0; f < FRAMES_PER_WAVE; ++f) {
    const int m = frame0 + wave * FRAMES_PER_WAVE + f;
    if (m >= M) break;
    float* orow = out + ((size_t)b * M + m) * (LR_C + 1);

    if (m >= total) {
      // Past this sequence's total duration: zeros (matches reference init).
#pragma unroll
      for (int kk = 0; kk < LR_C / 32; ++kk)
        __builtin_nontemporal_store(0.0f, orow + kk * 32 + lane);
      if (lane == 0) __builtin_nontemporal_store(0.0f, orow + LR_C);
      continue;
    }

    // searchsorted(cs, m, 'right'): first j with cs[j] > m. Uniform across
    // the wave (same m for all lanes) -> broadcast LDS reads, no divergence.
    int lo = 0, hi = LR_T;
    while (lo < hi) {
      int mid = (lo + hi) >> 1;
      if (s_cs[mid] <= m) lo = mid + 1; else hi = mid;
    }
    const int j = lo;  // m < total guarantees j <= T-1

    const int csj = s_cs[j];
    const int prev = (j > 0) ? s_cs[j - 1] : 0;  // token start frame
    const int d = csj - prev;                    // token duration (>=1 here)
    const float pos = (d > 1) ? (float)(m - prev) / (float)(d - 1) : 0.0f;

    const float* xrow = x + ((size_t)b * LR_T + j) * LR_C;
    // 12 coalesced dword loads (L2-resident x) + 12 coalesced NT dword stores.
#pragma unroll
    for (int kk = 0; kk < LR_C / 32; ++kk) {
      const int k = kk * 32 + lane;
      __builtin_nontemporal_store(xrow[k], orow + k);
    }
    if (lane == 0) __builtin_nontemporal_store(pos, orow + LR_C);
  }
}

// ---------------------------------------------------------------------------
extern "C" void kernel_launch(void* const* d_in, const int* in_sizes, int n_in,
                              void* d_out, int out_size, void* d_ws, size_t ws_size,
                              hipStream_t stream) {
  const float* x = (const float*)d_in[0];
  const int* dur = (const int*)d_in[1];
  float* out = (float*)d_out;
  int* cs = (int*)d_ws;  // 16*2048*4 = 128 KB scratch for cumsums

  const int M = out_size / (LR_B * (LR_C + 1));

  lr_cumsum_kernel<<<LR_B, 256, 0, stream>>>(dur, cs);

  dim3 grid((M + FRAMES_PER_BLOCK - 1) / FRAMES_PER_BLOCK, LR_B);
  lr_expand_kernel<<<grid, 256, 0, stream>>>(x, cs, out, M);
}